// LengthRegulator_22024592294513
// MI455X (gfx1250) — compile-verified
//
#include <hip/hip_runtime.h>
#include <hip/hip_bf16.h>
#include <stdint.h>

#define BATCH 16
#define T 512
#define D 384
#define D4 (D / 4)            // 96 float4 per feature row
#define ROWS_PER_BLOCK 8
#define GATHER_THREADS 96     // == D4: one float4 per thread per row (3 wave32s)

typedef __attribute__((ext_vector_type(4))) float f32x4;

// ---------------------------------------------------------------------------
// Kernel 1: per-batch inclusive scan of durations.
// One 512-thread block per batch; Hillis-Steele scan in LDS (9 steps).
// ---------------------------------------------------------------------------
__global__ void lr_cumsum_kernel(const int* __restrict__ durations,
                                 int* __restrict__ cum) {
  __shared__ int s[T];
  const int b = blockIdx.x;
  const int t = threadIdx.x;

  s[t] = durations[b * T + t];
  __syncthreads();

#pragma unroll
  for (int off = 1; off < T; off <<= 1) {
    const int v = (t >= off) ? s[t - off] : 0;
    __syncthreads();
    s[t] += v;
    __syncthreads();
  }
  cum[b * T + t] = s[t];
}

// ---------------------------------------------------------------------------
// Kernel 2: gather + expand.
// Block = (row-group p0..p0+7, batch b). Stages cum[b,:] (2 KB) into LDS via
// CDNA5 async global->LDS b128 copies (ASYNCcnt path), binary-searches the
// source frame for each of the 8 output rows in LDS, then streams the rows
// out with 128-bit non-temporal stores (output is write-once; keep x in L2).
// ---------------------------------------------------------------------------
__global__ void lr_gather_kernel(const float* __restrict__ x,
                                 const int* __restrict__ cum,
                                 float* __restrict__ out,
                                 int max_len) {
  __shared__ __align__(16) int scum[T];
  __shared__ int sidx[ROWS_PER_BLOCK];

  const int b   = blockIdx.y;
  const int p0  = blockIdx.x * ROWS_PER_BLOCK;
  const int tid = threadIdx.x;

  // --- async-stage cum[b, 0:512] into LDS: 128 chunks of 16 B ---
  {
    const int* cb = cum + b * T;
    const uint32_t lds_base = (uint32_t)(uintptr_t)(&scum[0]);  // low 32 bits = LDS byte offset
    for (int c = tid; c < T / 4; c += GATHER_THREADS) {
      const uint64_t gaddr = (uint64_t)(uintptr_t)(cb + 4 * c);
      const uint32_t laddr = lds_base + 16u * (uint32_t)c;
      asm volatile("global_load_async_to_lds_b128 %0, %1, off"
                   :
                   : "v"(laddr), "v"(gaddr)
                   : "memory");
    }
#if __has_builtin(__builtin_amdgcn_s_wait_asynccnt)
    __builtin_amdgcn_s_wait_asynccnt(0);
#else
    asm volatile("s_wait_asynccnt 0" ::: "memory");
#endif
  }
  __syncthreads();

  const int total = scum[T - 1];

  // --- searchsorted(cum, p, side='right'), one row per thread ---
  if (tid < ROWS_PER_BLOCK) {
    const int p = p0 + tid;
    int lo = 0, hi = T;
    while (lo < hi) {
      const int mid = (lo + hi) >> 1;
      if (scum[mid] <= p) lo = mid + 1;
      else                hi = mid;
    }
    sidx[tid] = (lo < T - 1) ? lo : (T - 1);   // jnp.minimum(idx, T-1)
  }
  __syncthreads();

  // --- copy 8 rows, one float4 per thread per row, NT stores ---
  const f32x4 zero4 = (f32x4)(0.0f);
#pragma unroll
  for (int r = 0; r < ROWS_PER_BLOCK; ++r) {
    const int p = p0 + r;
    if (p >= max_len) break;
    const bool valid = (p < total);
    const f32x4* __restrict__ src4 =
        (const f32x4*)(x + ((size_t)b * T + sidx[r]) * D);
    f32x4* __restrict__ dst4 =
        (f32x4*)(out + ((size_t)b * max_len + p) * D);
    const f32x4 v = valid ? src4[tid] : zero4;
    __builtin_nontemporal_store(v, dst4 + tid);
  }
}

// ---------------------------------------------------------------------------
extern "C" void kernel_launch(void* const* d_in, const int* in_sizes, int n_in,
                              void* d_out, int out_size, void* d_ws, size_t ws_size,
                              hipStream_t stream) {
  const float* x         = (const float*)d_in[0];   // (16, 512, 384) fp32
  const int*   durations = (const int*)d_in[1];     // (16, 512) int
  float*       out       = (float*)d_out;           // (16, max_len, 384) fp32
  int*         cum       = (int*)d_ws;              // 16*512 ints = 32 KB scratch

  const int max_len = out_size / (BATCH * D);

  lr_cumsum_kernel<<<dim3(BATCH), dim3(T), 0, stream>>>(durations, cum);

  dim3 grid((max_len + ROWS_PER_BLOCK - 1) / ROWS_PER_BLOCK, BATCH);
  lr_gather_kernel<<<grid, dim3(GATHER_THREADS), 0, stream>>>(x, cum, out, max_len);
}